// WindowAttentionWithRelPos_91336774517015
// MI455X (gfx1250) — compile-verified
//
#include <hip/hip_runtime.h>
#include <hip/hip_bf16.h>
#include <math.h>

// ---------------- problem constants ----------------
#define Bsz 4
#define Hdim 96
#define Wdim 96
#define Cdim 192
#define HEADS 6
#define WIN 16
#define SHIFT 8
#define HD 32
#define Ntok 256
#define NWX 6
#define NW 36
#define NB (Bsz * NW)                            // 144 windows total
#define QSEG ((size_t)NB * HEADS * Ntok * HD)    // 7,077,888 floats per q/k/v segment
#define OSEG ((size_t)NB * Ntok * Cdim)          // 7,077,888 floats

typedef __attribute__((ext_vector_type(2))) float v2f;
typedef __attribute__((ext_vector_type(8))) float v8f;

__device__ __forceinline__ v8f wmma_f32_4(v2f a, v2f b, v8f c) {
  // D = A(16x4 f32) * B(4x16 f32) + C(16x16 f32)
  return __builtin_amdgcn_wmma_f32_16x16x4_f32(false, a, false, b, (short)0, c,
                                               false, false);
}

// async global->LDS copy of 16 bytes per lane (ASYNCcnt-tracked)
__device__ __forceinline__ void async_copy_b128(unsigned lds_byte_off,
                                                const float* gaddr) {
  asm volatile("global_load_async_to_lds_b128 %0, %1, off"
               :
               : "v"(lds_byte_off), "v"(gaddr)
               : "memory");
}
__device__ __forceinline__ void wait_async0() {
  asm volatile("s_wait_asynccnt 0x0" ::: "memory");
}

// ---------------- kernel 1: CPB MLP -> bias_table (961 x 6) ----------------
__global__ void cpb_mlp_kernel(const float* __restrict__ tab,   // (961,2)
                               const float* __restrict__ w1,    // (512,2)
                               const float* __restrict__ b1,    // (512)
                               const float* __restrict__ w2,    // (6,512)
                               float* __restrict__ bias_table)  // (961,6)
{
  int g = blockIdx.x * blockDim.x + threadIdx.x;
  if (g >= 961 * HEADS) return;
  int p = g / HEADS, h = g % HEADS;
  float t0 = tab[p * 2 + 0], t1 = tab[p * 2 + 1];
  float acc = 0.f;
  for (int j = 0; j < 512; ++j) {
    float hj = fmaxf(w1[j * 2 + 0] * t0 + w1[j * 2 + 1] * t1 + b1[j], 0.f);
    acc += w2[h * 512 + j] * hj;
  }
  bias_table[g] = acc;  // row-major (p, h)
}

// ---------------- kernel 2: QKV GEMM (fused shift + window partition) ------
// out = xw @ qkv_w^T, M=36864, N=576, K=192.
// One wave computes a 16x64 strip (4 accumulators, A fragment reused 4x).
__global__ void qkv_gemm_kernel(const float* __restrict__ x,      // (B,96,96,192)
                                const float* __restrict__ qkv_w,  // (576,192)
                                const float* __restrict__ q_bias, // (192)
                                const float* __restrict__ v_bias, // (192)
                                float* __restrict__ qkv)          // 3 * QSEG
{
  const int tid = threadIdx.x;
  const int lane = tid & 31, wid = tid >> 5;
  const int l15 = lane & 15, kh = lane >> 4;

  int grp = blockIdx.x * 8 + wid;            // 2304 * 9 = 20736 strips
  int tM = grp / 9, gN = grp % 9;            // gN over 576/64

  // A row pointer: constant over K loop (shift+window gather folded in)
  int m = tM * 16 + l15;
  int bwin = m >> 8, n = m & 255;
  int b = bwin / NW, win = bwin % NW;
  int wy = win / NWX, wx = win % NWX;
  int r = n >> 4, c = n & 15;
  int hs = wy * WIN + r + SHIFT; if (hs >= Hdim) hs -= Hdim;
  int ws = wx * WIN + c + SHIFT; if (ws >= Wdim) ws -= Wdim;
  const float* arow = x + ((size_t)(b * Hdim + hs) * Wdim + ws) * Cdim;

  const float* brow[4];
#pragma unroll
  for (int t = 0; t < 4; ++t)
    brow[t] = qkv_w + (size_t)(gN * 64 + t * 16 + l15) * Cdim;

  v8f acc[4];
#pragma unroll
  for (int t = 0; t < 4; ++t) acc[t] = (v8f){0.f,0.f,0.f,0.f,0.f,0.f,0.f,0.f};

#pragma unroll 4
  for (int k0 = 0; k0 < Cdim; k0 += 4) {
    v2f a;
    a.x = arow[k0 + 2 * kh];
    a.y = arow[k0 + 2 * kh + 1];
#pragma unroll
    for (int t = 0; t < 4; ++t) {
      v2f bb;
      bb.x = brow[t][k0 + 2 * kh];
      bb.y = brow[t][k0 + 2 * kh + 1];
      acc[t] = wmma_f32_4(a, bb, acc[t]);
    }
  }

  // epilogue: bias + scatter into (three, bwin, head, n, d)
#pragma unroll
  for (int t = 0; t < 4; ++t) {
    int f = gN * 64 + t * 16 + l15;
    int three = f / Cdim, fh = f % Cdim;
    int head = fh >> 5, d = fh & 31;
    float bias = 0.f;
    if (three == 0) bias = q_bias[fh];
    else if (three == 2) bias = v_bias[fh];
    float* seg = qkv + (size_t)three * QSEG;
#pragma unroll
    for (int i = 0; i < 8; ++i) {
      int mo = tM * 16 + i + 8 * kh;
      int bw2 = mo >> 8, n2 = mo & 255;
      seg[(((size_t)bw2 * HEADS + head) * Ntok + n2) * HD + d] = acc[t][i] + bias;
    }
  }
}

// ---------------- kernel 3: attention (one block per window,head,qblock) ---
// LDS: sQ 64x36 | sK 256x36 | sV 256x36 | sS 64x260  = 149,504 bytes
#define QB 64
#define KV_STRIDE 36          // 144B rows: 16B-aligned for async b128, conflict-free
#define SS_STRIDE 260
#define ATTN_LDS_FLOATS (QB * KV_STRIDE + 2 * Ntok * KV_STRIDE + QB * SS_STRIDE)
#define ATTN_LDS_BYTES (ATTN_LDS_FLOATS * 4)

__global__ void attn_kernel(const float* __restrict__ qkv,         // 3*QSEG
                            const float* __restrict__ logit_scale, // (6)
                            const float* __restrict__ bias_table,  // (961,6)
                            const int* __restrict__ rel_idx,       // (256,256)
                            const float* __restrict__ mask,        // (36,1,256,256)
                            float* __restrict__ ohead)             // (NB,N,192)
{
  extern __shared__ float lds[];
  float* sQ = lds;
  float* sK = sQ + QB * KV_STRIDE;
  float* sV = sK + Ntok * KV_STRIDE;
  float* sS = sV + Ntok * KV_STRIDE;

  const unsigned sQ_off = (unsigned)(size_t)sQ;   // flat->LDS: low 32 bits
  const unsigned sK_off = (unsigned)(size_t)sK;
  const unsigned sV_off = (unsigned)(size_t)sV;

  const int tid = threadIdx.x;
  const int bx = blockIdx.x;
  const int qb = bx & 3;
  const int hb = (bx >> 2) % HEADS;
  const int bwin = bx / (4 * HEADS);
  const int w = bwin % NW;

  const float* qbase = qkv + ((size_t)bwin * HEADS + hb) * Ntok * HD;
  const float* kbase = qbase + QSEG;
  const float* vbase = qbase + 2 * QSEG;

  // ---- async DMA K, V (256x32 f32) and Q block (64x32) into LDS ----
  // rows are contiguous 128B in global; 16B chunk g -> row g>>3, chunk g&7
#pragma unroll
  for (int g = tid; g < 2048; g += 256) {       // K: 32KB
    unsigned dst = sK_off + (g >> 3) * (KV_STRIDE * 4) + (g & 7) * 16;
    async_copy_b128(dst, kbase + g * 4);
  }
#pragma unroll
  for (int g = tid; g < 2048; g += 256) {       // V: 32KB
    unsigned dst = sV_off + (g >> 3) * (KV_STRIDE * 4) + (g & 7) * 16;
    async_copy_b128(dst, vbase + g * 4);
  }
#pragma unroll
  for (int g = tid; g < 512; g += 256) {        // Q block: 8KB
    unsigned dst = sQ_off + (g >> 3) * (KV_STRIDE * 4) + (g & 7) * 16;
    async_copy_b128(dst, qbase + (size_t)qb * QB * HD + g * 4);
  }
  wait_async0();
  __syncthreads();

  // ---- in-place cosine normalization (K rows: all threads; Q rows: tid<64)
  {
    float ss = 0.f;
    float* krow = sK + tid * KV_STRIDE;
#pragma unroll
    for (int j = 0; j < HD; ++j) ss += krow[j] * krow[j];
    float inv = 1.f / (sqrtf(ss) + 1e-12f);
#pragma unroll
    for (int j = 0; j < HD; ++j) krow[j] *= inv;
  }
  if (tid < QB) {
    float ss = 0.f;
    float* qrow = sQ + tid * KV_STRIDE;
#pragma unroll
    for (int j = 0; j < HD; ++j) ss += qrow[j] * qrow[j];
    float inv = 1.f / (sqrtf(ss) + 1e-12f);
#pragma unroll
    for (int j = 0; j < HD; ++j) qrow[j] *= inv;
  }
  __syncthreads();

  const float scale = __expf(fminf(logit_scale[hb], 4.6051701859880914f));
  const int lane = tid & 31, wid = tid >> 5;
  const int l15 = lane & 15, kh = lane >> 4;

  // ---- S = scale*(Qn Kn^T) + rpb + mask ----
  // wave owns tm = wid>>1 and 8 tn tiles; Q fragment reused 8x per k-step.
  {
    int tm = wid >> 1;
    int tn0 = (wid & 1) * 8;
    const float* qrow = sQ + (tm * 16 + l15) * KV_STRIDE;
    v8f acc[8];
#pragma unroll
    for (int j = 0; j < 8; ++j) acc[j] = (v8f){0.f,0.f,0.f,0.f,0.f,0.f,0.f,0.f};
#pragma unroll
    for (int k0 = 0; k0 < HD; k0 += 4) {
      v2f a;
      a.x = qrow[k0 + 2 * kh];
      a.y = qrow[k0 + 2 * kh + 1];
#pragma unroll
      for (int j = 0; j < 8; ++j) {
        const float* krow = sK + ((tn0 + j) * 16 + l15) * KV_STRIDE;
        v2f b;
        b.x = krow[k0 + 2 * kh];
        b.y = krow[k0 + 2 * kh + 1];
        acc[j] = wmma_f32_4(a, b, acc[j]);
      }
    }
#pragma unroll
    for (int j = 0; j < 8; ++j) {
      int col = (tn0 + j) * 16 + l15;
#pragma unroll
      for (int i = 0; i < 8; ++i) {
        int row = tm * 16 + i + 8 * kh;
        int nq = qb * QB + row;
        int ridx = rel_idx[nq * Ntok + col];
        float bt = bias_table[ridx * HEADS + hb];
        float rpb = 16.f / (1.f + __expf(-bt));
        float mk = mask[((size_t)w * Ntok + nq) * Ntok + col];
        sS[row * SS_STRIDE + col] = acc[j][i] * scale + rpb + mk;
      }
    }
  }
  __syncthreads();

  // ---- softmax over rows of 256; each wave handles 8 rows ----
  for (int rr = 0; rr < 8; ++rr) {
    int row = wid * 8 + rr;
    float* srow = sS + row * SS_STRIDE;
    float vals[8];
    float mx = -1e30f;
#pragma unroll
    for (int j = 0; j < 8; ++j) { vals[j] = srow[lane + j * 32]; mx = fmaxf(mx, vals[j]); }
    for (int off = 16; off > 0; off >>= 1) mx = fmaxf(mx, __shfl_xor(mx, off, 32));
    float sum = 0.f;
#pragma unroll
    for (int j = 0; j < 8; ++j) { vals[j] = __expf(vals[j] - mx); sum += vals[j]; }
    for (int off = 16; off > 0; off >>= 1) sum += __shfl_xor(sum, off, 32);
    float inv = 1.f / sum;
#pragma unroll
    for (int j = 0; j < 8; ++j) srow[lane + j * 32] = vals[j] * inv;
  }
  __syncthreads();

  // ---- O = P @ V  (8 tiles of 16x16, one per wave) ----
  {
    int tm = wid >> 1, tn = wid & 1;
    const float* prow = sS + (tm * 16 + l15) * SS_STRIDE;
    v8f acc = {0.f, 0.f, 0.f, 0.f, 0.f, 0.f, 0.f, 0.f};
#pragma unroll 8
    for (int k0 = 0; k0 < Ntok; k0 += 4) {
      v2f a, b;
      a.x = prow[k0 + 2 * kh];  a.y = prow[k0 + 2 * kh + 1];
      b.x = sV[(k0 + 2 * kh) * KV_STRIDE + tn * 16 + l15];
      b.y = sV[(k0 + 2 * kh + 1) * KV_STRIDE + tn * 16 + l15];
      acc = wmma_f32_4(a, b, acc);
    }
    int d = tn * 16 + l15;
#pragma unroll
    for (int i = 0; i < 8; ++i) {
      int row = tm * 16 + i + 8 * kh;
      int nq = qb * QB + row;
      ohead[((size_t)bwin * Ntok + nq) * Cdim + hb * HD + d] = acc[i];
    }
  }
}

// ---------------- kernel 4: proj GEMM + reverse window/shift ---------------
// M=36864, N=192, K=192; one wave per 16x64 strip (4 accumulators).
__global__ void proj_gemm_kernel(const float* __restrict__ ohead,  // (M,192)
                                 const float* __restrict__ proj_w, // (192,192)
                                 const float* __restrict__ proj_b, // (192)
                                 float* __restrict__ out)          // (B,96,96,192)
{
  const int tid = threadIdx.x;
  const int lane = tid & 31, wid = tid >> 5;
  const int l15 = lane & 15, kh = lane >> 4;

  int grp = blockIdx.x * 8 + wid;       // 2304 * 3 = 6912 strips
  int tM = grp / 3, gN = grp % 3;

  int m = tM * 16 + l15;
  const float* arow = ohead + (size_t)m * Cdim;
  const float* brow[4];
#pragma unroll
  for (int t = 0; t < 4; ++t)
    brow[t] = proj_w + (size_t)(gN * 64 + t * 16 + l15) * Cdim;

  v8f acc[4];
#pragma unroll
  for (int t = 0; t < 4; ++t) acc[t] = (v8f){0.f,0.f,0.f,0.f,0.f,0.f,0.f,0.f};

#pragma unroll 4
  for (int k0 = 0; k0 < Cdim; k0 += 4) {
    v2f a;
    a.x = arow[k0 + 2 * kh];
    a.y = arow[k0 + 2 * kh + 1];
#pragma unroll
    for (int t = 0; t < 4; ++t) {
      v2f bb;
      bb.x = brow[t][k0 + 2 * kh];
      bb.y = brow[t][k0 + 2 * kh + 1];
      acc[t] = wmma_f32_4(a, bb, acc[t]);
    }
  }

#pragma unroll
  for (int t = 0; t < 4; ++t) {
    int f = gN * 64 + t * 16 + l15;
    float pb = proj_b[f];
#pragma unroll
    for (int i = 0; i < 8; ++i) {
      int mo = tM * 16 + i + 8 * kh;
      int bwin = mo >> 8, n = mo & 255;
      int b = bwin / NW, win = bwin % NW;
      int wy = win / NWX, wx = win % NWX;
      int r = n >> 4, c = n & 15;
      int hd = wy * WIN + r + SHIFT; if (hd >= Hdim) hd -= Hdim;
      int wd = wx * WIN + c + SHIFT; if (wd >= Wdim) wd -= Wdim;
      out[((size_t)(b * Hdim + hd) * Wdim + wd) * Cdim + f] = acc[t][i] + pb;
    }
  }
}

// ---------------- launch ----------------
extern "C" void kernel_launch(void* const* d_in, const int* in_sizes, int n_in,
                              void* d_out, int out_size, void* d_ws, size_t ws_size,
                              hipStream_t stream) {
  const float* x            = (const float*)d_in[0];
  const float* qkv_w        = (const float*)d_in[1];
  const float* q_bias       = (const float*)d_in[2];
  const float* v_bias       = (const float*)d_in[3];
  const float* logit_scale  = (const float*)d_in[4];
  const float* cpb_w1       = (const float*)d_in[5];
  const float* cpb_b1       = (const float*)d_in[6];
  const float* cpb_w2       = (const float*)d_in[7];
  const float* proj_w       = (const float*)d_in[8];
  const float* proj_b       = (const float*)d_in[9];
  const float* rel_tab      = (const float*)d_in[10];
  const int*   rel_idx      = (const int*)d_in[11];
  const float* attn_mask    = (const float*)d_in[12];
  float* out = (float*)d_out;

  float* ws    = (float*)d_ws;
  float* qkv   = ws;                       // 3 * QSEG floats
  float* ohead = ws + 3 * QSEG;            // OSEG floats
  float* btab  = ohead + OSEG;             // 961*6 floats

  // 1) CPB MLP -> bias table
  cpb_mlp_kernel<<<(961 * HEADS + 255) / 256, 256, 0, stream>>>(
      rel_tab, cpb_w1, cpb_b1, cpb_w2, btab);

  // 2) QKV projection (fused shift + window partition), 16x64 strips
  qkv_gemm_kernel<<<(2304 * 9) / 8, 256, 0, stream>>>(
      x, qkv_w, q_bias, v_bias, qkv);

  // 3) attention (149.5 KB dynamic LDS per workgroup; 320 KB on WGP)
  hipFuncSetAttribute(reinterpret_cast<const void*>(attn_kernel),
                      hipFuncAttributeMaxDynamicSharedMemorySize,
                      ATTN_LDS_BYTES);
  attn_kernel<<<NB * HEADS * (Ntok / QB), 256, ATTN_LDS_BYTES, stream>>>(
      qkv, logit_scale, btab, rel_idx, attn_mask, ohead);

  // 4) output projection + reverse window/shift, 16x64 strips
  proj_gemm_kernel<<<(2304 * 3) / 8, 256, 0, stream>>>(
      ohead, proj_w, proj_b, out);
}